// GATEncoderAdv_29085518528543
// MI455X (gfx1250) — compile-verified
//
#include <hip/hip_runtime.h>

#define N_NODES 40000
#define N_EDGES 640000
#define N_GRAPHS 128
#define N_IN 64
#define NHID 128
#define NOUT 256
#define NEG_SLOPE 0.2f

#define FLAG_RELU 1
#define FLAG_ADD  2

typedef float v2f __attribute__((ext_vector_type(2)));
typedef float v8f __attribute__((ext_vector_type(8)));

// ---------------------------------------------------------------- utilities
__global__ void fill_kernel(float* __restrict__ p, float v, int n) {
  int i = blockIdx.x * blockDim.x + threadIdx.x;
  if (i < n) p[i] = v;
}

// dst[i] = b[i % C]  (C power of two)
__global__ void init_bias_kernel(float* __restrict__ dst, const float* __restrict__ b,
                                 int n, int C) {
  int i = blockIdx.x * blockDim.x + threadIdx.x;
  if (i < n) dst[i] = b[i & (C - 1)];
}

// dst[i] = scale * max(src[i], 0)
__global__ void relu_scale_kernel(float* __restrict__ dst, const float* __restrict__ src,
                                  float scale, int n) {
  int i = blockIdx.x * blockDim.x + threadIdx.x;
  if (i < n) dst[i] = scale * fmaxf(src[i], 0.0f);
}

// ---------------------------------------------------------------- WMMA GEMM (single B)
// C[M,NCOL] = A[M,K] @ B[K,NCOL] (+bias) (relu?) (+addsrc?)
// One wave -> 16 rows x NCOL cols, f32 WMMA 16x16x4 (exact fp32 math).
// A-frag (16x4 f32, ISA 7.12.2): v0 = K{0,2}, v1 = K{1,3} across lane halves.
// B-frag (4x16): v0 = rows {0,2}, v1 = rows {1,3} across lane halves.
// C/D (16x16 f32): VGPR v -> rows v (lanes 0-15) and v+8 (lanes 16-31).
template <int NCOL>
__global__ __launch_bounds__(256) void gemm_wmma_kernel(
    const float* __restrict__ A, const float* __restrict__ B,
    const float* __restrict__ bias, const float* __restrict__ addsrc,
    float* __restrict__ C, int M, int K, int flags) {
  constexpr int NT = NCOL / 16;
  int wave = (int)((blockIdx.x * blockDim.x + threadIdx.x) >> 5);
  int lane = threadIdx.x & 31;
  int m0 = wave << 4;
  if (m0 >= M) return;                 // wave-uniform: EXEC stays all-ones
  int hl  = lane >> 4;                 // lane half
  int l16 = lane & 15;

  v8f acc[NT] = {};
  const float* arow = A + (size_t)(m0 + l16) * K;
  for (int k = 0; k < K; k += 4) {
    v2f a = *(const v2f*)(arow + k + 2 * hl);   // one b64 load
    const float* b0 = B + (size_t)(k + 2 * hl) * NCOL;
    const float* b1 = b0 + NCOL;
#pragma unroll
    for (int t = 0; t < NT; ++t) {
      int n = t * 16 + l16;
      v2f b;
      b.x = b0[n];
      b.y = b1[n];
      acc[t] = __builtin_amdgcn_wmma_f32_16x16x4_f32(
          false, a, false, b, (short)0, acc[t], false, false);
    }
  }
#pragma unroll
  for (int t = 0; t < NT; ++t) {
    int n = t * 16 + l16;
    float bv = bias ? bias[n] : 0.0f;
#pragma unroll
    for (int v = 0; v < 8; ++v) {
      int m = m0 + v + 8 * hl;
      float val = acc[t][v] + bv;
      if (flags & FLAG_RELU) val = fmaxf(val, 0.0f);
      if (flags & FLAG_ADD)  val += addsrc[(size_t)m * NCOL + n];
      C[(size_t)m * NCOL + n] = val;
    }
  }
}

// ---------------------------------------------------------------- WMMA GEMM (dual B)
// C1 = A @ B1, C2 = A @ B2 with one pass over A (A fragment reused across
// 2*NHID/16 = 16 N-tiles -> halves A global traffic for the Wl/Wr pair).
__global__ __launch_bounds__(256) void gemm_dual_wmma_kernel(
    const float* __restrict__ A,
    const float* __restrict__ B1, const float* __restrict__ B2,
    float* __restrict__ C1, float* __restrict__ C2, int M, int K) {
  constexpr int NT = NHID / 16;  // 8 tiles per output
  int wave = (int)((blockIdx.x * blockDim.x + threadIdx.x) >> 5);
  int lane = threadIdx.x & 31;
  int m0 = wave << 4;
  if (m0 >= M) return;
  int hl  = lane >> 4;
  int l16 = lane & 15;

  v8f acc1[NT] = {};
  v8f acc2[NT] = {};
  const float* arow = A + (size_t)(m0 + l16) * K;
  for (int k = 0; k < K; k += 4) {
    v2f a = *(const v2f*)(arow + k + 2 * hl);
    const float* p1 = B1 + (size_t)(k + 2 * hl) * NHID;
    const float* p2 = B2 + (size_t)(k + 2 * hl) * NHID;
#pragma unroll
    for (int t = 0; t < NT; ++t) {
      int n = t * 16 + l16;
      v2f b;
      b.x = p1[n];
      b.y = p1[n + NHID];
      acc1[t] = __builtin_amdgcn_wmma_f32_16x16x4_f32(
          false, a, false, b, (short)0, acc1[t], false, false);
    }
#pragma unroll
    for (int t = 0; t < NT; ++t) {
      int n = t * 16 + l16;
      v2f b;
      b.x = p2[n];
      b.y = p2[n + NHID];
      acc2[t] = __builtin_amdgcn_wmma_f32_16x16x4_f32(
          false, a, false, b, (short)0, acc2[t], false, false);
    }
  }
#pragma unroll
  for (int t = 0; t < NT; ++t) {
    int n = t * 16 + l16;
#pragma unroll
    for (int v = 0; v < 8; ++v) {
      int m = m0 + v + 8 * hl;
      C1[(size_t)m * NHID + n] = acc1[t][v];
      C2[(size_t)m * NHID + n] = acc2[t][v];
    }
  }
}

// ---------------------------------------------------------------- edge kernels
__device__ __forceinline__ float lrelu(float x) {
  return x > 0.0f ? x : NEG_SLOPE * x;
}

// wave per edge: e = att . lrelu(xl[src] + xr[dst]); segment max via f32 atomic max
__global__ void edge_score_kernel(const float* __restrict__ xl, const float* __restrict__ xr,
                                  const int* __restrict__ src, const int* __restrict__ dst,
                                  const float* __restrict__ att,
                                  float* __restrict__ escore, float* __restrict__ mmax) {
  int edge = (int)((blockIdx.x * blockDim.x + threadIdx.x) >> 5);
  int lane = threadIdx.x & 31;
  if (edge >= N_EDGES) return;
  int s = src[edge], d = dst[edge];
  float4 l = ((const float4*)(xl + (size_t)s * NHID))[lane];
  float4 r = ((const float4*)(xr + (size_t)d * NHID))[lane];
  float4 a = ((const float4*)att)[lane];
  float sum = a.x * lrelu(l.x + r.x) + a.y * lrelu(l.y + r.y) +
              a.z * lrelu(l.z + r.z) + a.w * lrelu(l.w + r.w);
#pragma unroll
  for (int off = 16; off > 0; off >>= 1) sum += __shfl_xor(sum, off, 32);
  if (lane == 0) {
    escore[edge] = sum;
    __hip_atomic_fetch_max(&mmax[d], sum, __ATOMIC_RELAXED, __HIP_MEMORY_SCOPE_AGENT);
  }
}

// thread per edge: ex = exp(e - m[dst]); segment sum via atomic add
__global__ void edge_exp_kernel(const float* __restrict__ escore, const int* __restrict__ dst,
                                const float* __restrict__ mmax, float* __restrict__ ssum,
                                float* __restrict__ ex) {
  int e = blockIdx.x * blockDim.x + threadIdx.x;
  if (e >= N_EDGES) return;
  int d = dst[e];
  float v = expf(escore[e] - mmax[d]);
  ex[e] = v;
  atomicAdd(&ssum[d], v);
}

// wave per edge: agg[dst] += alpha * xl[src] (128-wide f32 atomic scatter)
__global__ void edge_scatter_kernel(const float* __restrict__ xl, const int* __restrict__ src,
                                    const int* __restrict__ dst, const float* __restrict__ ex,
                                    const float* __restrict__ ssum, float* __restrict__ agg) {
  int edge = (int)((blockIdx.x * blockDim.x + threadIdx.x) >> 5);
  int lane = threadIdx.x & 31;
  if (edge >= N_EDGES) return;
  int s = src[edge], d = dst[edge];
  float alpha = ex[edge] / (ssum[d] + 1e-16f);
  float4 l = ((const float4*)(xl + (size_t)s * NHID))[lane];
  float* o = agg + (size_t)d * NHID + lane * 4;
  atomicAdd(o + 0, alpha * l.x);
  atomicAdd(o + 1, alpha * l.y);
  atomicAdd(o + 2, alpha * l.z);
  atomicAdd(o + 3, alpha * l.w);
}

// ---------------------------------------------------------------- pooling
__global__ void pool_sum_kernel(const float* __restrict__ h, const int* __restrict__ batch,
                                float* __restrict__ pooled, float* __restrict__ cnt) {
  int node = (int)((blockIdx.x * blockDim.x + threadIdx.x) >> 5);
  int lane = threadIdx.x & 31;
  if (node >= N_NODES) return;
  int g = batch[node];
  float4 v = ((const float4*)(h + (size_t)node * NHID))[lane];
  float* o = pooled + (size_t)g * NHID + lane * 4;
  atomicAdd(o + 0, v.x);
  atomicAdd(o + 1, v.y);
  atomicAdd(o + 2, v.z);
  atomicAdd(o + 3, v.w);
  if (lane == 0) atomicAdd(&cnt[g], 1.0f);
}

__global__ void pool_div_kernel(float* __restrict__ pooled, const float* __restrict__ cnt) {
  int i = blockIdx.x * blockDim.x + threadIdx.x;
  if (i >= N_GRAPHS * NHID) return;
  pooled[i] /= fmaxf(cnt[i / NHID], 1.0f);
}

// ---------------------------------------------------------------- layernorm
// wave per graph, 256 dims -> 8 elems/lane
__global__ void layernorm_kernel(const float* __restrict__ z, const float* __restrict__ g,
                                 const float* __restrict__ b, float* __restrict__ out) {
  int gr = (int)((blockIdx.x * blockDim.x + threadIdx.x) >> 5);
  int lane = threadIdx.x & 31;
  if (gr >= N_GRAPHS) return;
  float v[8];
  float sum = 0.0f;
#pragma unroll
  for (int i = 0; i < 8; ++i) {
    v[i] = z[(size_t)gr * NOUT + i * 32 + lane];
    sum += v[i];
  }
#pragma unroll
  for (int off = 16; off > 0; off >>= 1) sum += __shfl_xor(sum, off, 32);
  float mean = sum * (1.0f / NOUT);
  float var = 0.0f;
#pragma unroll
  for (int i = 0; i < 8; ++i) {
    float d = v[i] - mean;
    var += d * d;
  }
#pragma unroll
  for (int off = 16; off > 0; off >>= 1) var += __shfl_xor(var, off, 32);
  var *= (1.0f / NOUT);
  float rstd = rsqrtf(var + 1e-5f);
#pragma unroll
  for (int i = 0; i < 8; ++i) {
    int c = i * 32 + lane;
    out[(size_t)gr * NOUT + c] = (v[i] - mean) * rstd * g[c] + b[c];
  }
}

// ---------------------------------------------------------------- launcher
extern "C" void kernel_launch(void* const* d_in, const int* in_sizes, int n_in,
                              void* d_out, int out_size, void* d_ws, size_t ws_size,
                              hipStream_t stream) {
  const float* x     = (const float*)d_in[0];
  const int*   ei    = (const int*)d_in[1];
  const int*   src   = ei;
  const int*   dstp  = ei + N_EDGES;
  const int*   batch = (const int*)d_in[2];
  const float* Wl[4]  = {(const float*)d_in[3],  (const float*)d_in[7],
                         (const float*)d_in[11], (const float*)d_in[15]};
  const float* Wr[4]  = {(const float*)d_in[4],  (const float*)d_in[8],
                         (const float*)d_in[12], (const float*)d_in[16]};
  const float* att[4] = {(const float*)d_in[5],  (const float*)d_in[9],
                         (const float*)d_in[13], (const float*)d_in[17]};
  const float* bia[4] = {(const float*)d_in[6],  (const float*)d_in[10],
                         (const float*)d_in[14], (const float*)d_in[18]};
  const float* res_w0 = (const float*)d_in[19];
  const float* res_b0 = (const float*)d_in[20];
  const float* res_w2 = (const float*)d_in[21];
  const float* res_b2 = (const float*)d_in[22];
  const float* mh1_w  = (const float*)d_in[23];
  const float* mh1_b  = (const float*)d_in[24];
  const float* mh2_w  = (const float*)d_in[25];
  const float* mh2_b  = (const float*)d_in[26];
  const float* ln_g   = (const float*)d_in[27];
  const float* ln_b   = (const float*)d_in[28];

  float* ws = (float*)d_ws;
  size_t off = 0;
  auto alloc = [&](size_t n) { float* p = ws + off; off += n; return p; };
  float* h      = alloc((size_t)N_NODES * NHID);
  float* xl     = alloc((size_t)N_NODES * NHID);
  float* xr     = alloc((size_t)N_NODES * NHID);
  float* agg    = alloc((size_t)N_NODES * NHID);
  float* esc    = alloc(N_EDGES);
  float* ex     = alloc(N_EDGES);
  float* mmax   = alloc(N_NODES);
  float* ssum   = alloc(N_NODES);
  float* pooled = alloc((size_t)N_GRAPHS * NHID);
  float* cnt    = alloc(N_GRAPHS);
  float* z1     = alloc((size_t)N_GRAPHS * NOUT);
  float* z      = alloc((size_t)N_GRAPHS * NOUT);

  const int TB = 256;
  const int NH = N_NODES * NHID;
  int gemmNodeBlocks = ((N_NODES / 16) * 32 + TB - 1) / TB;  // wave per 16-row strip
  int edgeWaveBlocks = (N_EDGES * 32 + TB - 1) / TB;         // wave per edge
  int nodeWaveBlocks = (N_NODES * 32 + TB - 1) / TB;         // wave per node
  int nhBlocks       = (NH + TB - 1) / TB;

  const float* hin = x;
  int K = N_IN;
  for (int l = 0; l < 4; ++l) {
    gemm_dual_wmma_kernel<<<gemmNodeBlocks, TB, 0, stream>>>(
        hin, Wl[l], Wr[l], xl, xr, N_NODES, K);
    init_bias_kernel<<<nhBlocks, TB, 0, stream>>>(agg, bia[l], NH, NHID);
    fill_kernel<<<(N_NODES + TB - 1) / TB, TB, 0, stream>>>(mmax, -3.0e38f, N_NODES);
    fill_kernel<<<(N_NODES + TB - 1) / TB, TB, 0, stream>>>(ssum, 0.0f, N_NODES);
    edge_score_kernel<<<edgeWaveBlocks, TB, 0, stream>>>(xl, xr, src, dstp, att[l], esc, mmax);
    edge_exp_kernel<<<(N_EDGES + TB - 1) / TB, TB, 0, stream>>>(esc, dstp, mmax, ssum, ex);
    edge_scatter_kernel<<<edgeWaveBlocks, TB, 0, stream>>>(xl, src, dstp, ex, ssum, agg);
    relu_scale_kernel<<<nhBlocks, TB, 0, stream>>>(agg, agg, 1.0f, NH);  // h_relu in agg
    if (l == 0) {
      gemm_wmma_kernel<NHID><<<gemmNodeBlocks, TB, 0, stream>>>(
          agg, res_w0, res_b0, agg, h, N_NODES, NHID, FLAG_ADD);
    } else if (l == 2) {
      gemm_wmma_kernel<NHID><<<gemmNodeBlocks, TB, 0, stream>>>(
          agg, res_w2, res_b2, agg, h, N_NODES, NHID, FLAG_ADD);
    } else {  // ReLU "residual": relu(h)+h == 2*h for h>=0
      relu_scale_kernel<<<nhBlocks, TB, 0, stream>>>(h, agg, 2.0f, NH);
    }
    hin = h;
    K = NHID;
  }

  fill_kernel<<<(N_GRAPHS * NHID + TB - 1) / TB, TB, 0, stream>>>(pooled, 0.0f, N_GRAPHS * NHID);
  fill_kernel<<<1, TB, 0, stream>>>(cnt, 0.0f, N_GRAPHS);
  pool_sum_kernel<<<nodeWaveBlocks, TB, 0, stream>>>(h, batch, pooled, cnt);
  pool_div_kernel<<<(N_GRAPHS * NHID + TB - 1) / TB, TB, 0, stream>>>(pooled, cnt);

  gemm_wmma_kernel<NOUT><<<1, TB, 0, stream>>>(
      pooled, mh1_w, mh1_b, nullptr, z1, N_GRAPHS, NHID, FLAG_RELU);
  gemm_wmma_kernel<NOUT><<<1, TB, 0, stream>>>(
      z1, mh2_w, mh2_b, nullptr, z, N_GRAPHS, NOUT, 0);
  layernorm_kernel<<<(N_GRAPHS * 32 + TB - 1) / TB, TB, 0, stream>>>(
      z, ln_g, ln_b, (float*)d_out);
}